// MLPPredictor_70153995813487
// MI455X (gfx1250) — compile-verified
//
#include <hip/hip_runtime.h>

typedef float v2f __attribute__((ext_vector_type(2)));
typedef float v8f __attribute__((ext_vector_type(8)));

#define D_DIM   128
#define E_DIM   64
#define OUT_DIM 2

// ---------------------------------------------------------------------------
// Phase 1: P4[n][0..1] = node_feats[n] @ Wu ; P4[n][2..3] = node_feats[n] @ Wv
// One wave computes a 16(nodes) x 16(cols, 4 used) tile via 32 x WMMA_F32_16X16X4_F32.
// ---------------------------------------------------------------------------
__global__ void __launch_bounds__(256) node_proj_wmma(
    const float* __restrict__ nf, const float* __restrict__ W,
    float* __restrict__ P4, int n_tiles)
{
    __shared__ float ldsB[D_DIM * 16];   // zero-padded [Wu | Wv | 0...] as 128x16
    for (int i = threadIdx.x; i < D_DIM * 16; i += 256) {
        int k = i >> 4, c = i & 15;
        float v = 0.0f;
        if (c < 2)      v = W[k * OUT_DIM + c];                    // Wu
        else if (c < 4) v = W[(D_DIM + k) * OUT_DIM + (c - 2)];    // Wv
        ldsB[i] = v;
    }
    __syncthreads();

    int wave = blockIdx.x * 8 + (threadIdx.x >> 5);
    if (wave >= n_tiles) return;                 // wave-uniform: EXEC stays all-1s
    int lane = threadIdx.x & 31;
    int row  = lane & 15;                        // A: M index; B: N index; C: N index
    int kof  = (lane >> 4) << 1;                 // 0 for lanes 0-15, 2 for lanes 16-31

    const float* arow = nf + (long)(wave * 16 + row) * D_DIM + kof;
    v8f acc = {};
    #pragma unroll 4
    for (int k = 0; k < D_DIM; k += 4) {
        v2f a = *(const v2f*)(arow + k);         // A: K={k+kof, k+kof+1} of row M
        v2f bmat;                                // B: K rows {k+kof, k+kof+1}, col N=row
        bmat.x = ldsB[(k + kof) * 16 + row];
        bmat.y = ldsB[(k + kof + 1) * 16 + row];
        acc = __builtin_amdgcn_wmma_f32_16x16x4_f32(false, a, false, bmat,
                                                    (short)0, acc, false, false);
    }

    // C layout: VGPR r -> M = r (+8 for high lanes), N = lane&15. Keep cols 0..3.
    if (row < 4) {
        int mbase = wave * 16 + ((lane >> 4) << 3);
        #pragma unroll
        for (int r = 0; r < 8; ++r)
            P4[(long)(mbase + r) * 4 + row] = acc[r];
    }
}

// ---------------------------------------------------------------------------
// Phase 2: out[e] = P4[src[e]][0..1] + P4[dst[e]][2..3] + edge_feats[e] @ We + b
// One wave handles 16 edges; edge dot via 16 x WMMA_F32_16X16X4_F32.
// Streams 409.6 MB of edge features; P4 gathers hit L2 (800 KB resident).
// ---------------------------------------------------------------------------
__global__ void __launch_bounds__(256) edge_score_wmma(
    const float* __restrict__ ef, const int* __restrict__ src,
    const int* __restrict__ dst, const float* __restrict__ W,
    const float* __restrict__ bias, const float* __restrict__ P4,
    float* __restrict__ outp, int n_tiles)
{
    __shared__ float ldsB[E_DIM * 16];   // zero-padded We as 64x16
    for (int i = threadIdx.x; i < E_DIM * 16; i += 256) {
        int k = i >> 4, c = i & 15;
        ldsB[i] = (c < 2) ? W[(2 * D_DIM + k) * OUT_DIM + c] : 0.0f;
    }
    __syncthreads();

    int wave = blockIdx.x * 8 + (threadIdx.x >> 5);
    if (wave >= n_tiles) return;                 // wave-uniform guard
    int lane = threadIdx.x & 31;
    int row  = lane & 15;
    int kof  = (lane >> 4) << 1;

    long ebase = (long)wave * 16;
    const float* arow = ef + (ebase + row) * E_DIM + kof;
    v8f acc = {};
    #pragma unroll
    for (int k = 0; k < E_DIM; k += 4) {
        v2f a = *(const v2f*)(arow + k);
        v2f bmat;
        bmat.x = ldsB[(k + kof) * 16 + row];
        bmat.y = ldsB[(k + kof + 1) * 16 + row];
        acc = __builtin_amdgcn_wmma_f32_16x16x4_f32(false, a, false, bmat,
                                                    (short)0, acc, false, false);
    }

    if (row < OUT_DIM) {                         // only cols 0,1 are real outputs
        float bv = bias[row];
        long mbase = ebase + ((lane >> 4) << 3);
        #pragma unroll
        for (int r = 0; r < 8; ++r) {
            long e = mbase + r;
            int s = src[e];
            int d = dst[e];
            outp[e * OUT_DIM + row] =
                acc[r] + P4[(long)s * 4 + row] + P4[(long)d * 4 + 2 + row] + bv;
        }
    }
}

// ---------------------------------------------------------------------------
extern "C" void kernel_launch(void* const* d_in, const int* in_sizes, int n_in,
                              void* d_out, int out_size, void* d_ws, size_t ws_size,
                              hipStream_t stream) {
    (void)n_in; (void)out_size; (void)ws_size;

    const float* nf   = (const float*)d_in[0];   // node_feats  [N_NODES, 128]
    const float* ef   = (const float*)d_in[1];   // edge_feats  [N_EDGES, 64]
    const int*   src  = (const int*)d_in[2];     // [N_EDGES]
    const int*   dst  = (const int*)d_in[3];     // [N_EDGES]
    const float* W    = (const float*)d_in[4];   // [320, 2] row-major
    const float* bias = (const float*)d_in[5];   // [2]
    float*       outp = (float*)d_out;           // [N_EDGES, 2]
    float*       P4   = (float*)d_ws;            // [N_NODES, 4] scratch

    int n_nodes = in_sizes[0] / D_DIM;           // 50000
    int n_edges = in_sizes[2];                   // 1600000
    int node_tiles = (n_nodes + 15) / 16;        // 3125 (exact)
    int edge_tiles = (n_edges + 15) / 16;        // 100000 (exact)

    node_proj_wmma<<<(node_tiles + 7) / 8, 256, 0, stream>>>(nf, W, P4, node_tiles);
    edge_score_wmma<<<(edge_tiles + 7) / 8, 256, 0, stream>>>(ef, src, dst, W, bias,
                                                              P4, outp, edge_tiles);
}